// AdditiveLowRankPairwise_29669634081456
// MI455X (gfx1250) — compile-verified
//
#include <hip/hip_runtime.h>
#include <hip/hip_bf16.h>

#define DIM 512
#define RANK 64
#define BATCH 2
#define TLEN 1024
#define SLEN 1024

typedef __attribute__((ext_vector_type(2))) float v2f;
typedef __attribute__((ext_vector_type(8))) float v8f;

// ---------------------------------------------------------------------------
// Kernel 1: low-rank projection P[m, r] = sum_d X[m, d] * W[r, d]
// One wave (32 threads) computes one 16x16 tile of P using
// V_WMMA_F32_16X16X4_F32, K-loop over DIM in steps of 4.
//
// VGPR layouts (CDNA5 ISA 7.12.2):
//   A 16x4 : lanes 0-15 -> M=lane,   vgpr j = K=j   ; lanes 16-31 -> K=j+2
//   B 4x16 : lanes 0-15 -> N=lane,   vgpr j = K=j   ; lanes 16-31 -> K=j+2
//   C 16x16: lanes 0-15 -> N=lane, vgpr j = M=j ; lanes 16-31 -> M=j+8
// ---------------------------------------------------------------------------
__global__ __launch_bounds__(32)
void proj_wmma_f32(const float* __restrict__ X,   // [M, DIM] row-major
                   const float* __restrict__ W,   // [RANK, DIM] row-major (torch-style)
                   float* __restrict__ P)         // [M, RANK] row-major
{
    const int lane  = threadIdx.x;      // 0..31
    const int half  = lane >> 4;        // 0 or 1
    const int l16   = lane & 15;
    const int mBase = blockIdx.x * 16;
    const int nBase = blockIdx.y * 16;

    const float* arow = X + (size_t)(mBase + l16) * DIM;  // A row for this lane
    const float* brow = W + (size_t)(nBase + l16) * DIM;  // B col r -> W row r

    v8f c = {};
    #pragma unroll 4
    for (int k0 = 0; k0 < DIM; k0 += 4) {
        const int kk = k0 + half * 2;
        v2f a, b;
        a.x = arow[kk];
        a.y = arow[kk + 1];
        b.x = brow[kk];
        b.y = brow[kk + 1];
        // 8 args: (neg_a, A, neg_b, B, c_mod, C, reuse_a, reuse_b)
        c = __builtin_amdgcn_wmma_f32_16x16x4_f32(
                false, a, false, b, (short)0, c, false, false);
    }

    const int col = nBase + l16;
    #pragma unroll
    for (int j = 0; j < 8; ++j) {
        const int row = mBase + j + half * 8;
        P[(size_t)row * RANK + col] = c[j];
    }
}

// ---------------------------------------------------------------------------
// Kernel 2: pairwise interaction
//   scores[b,t,s] = sum_r silu(pt[b,t,r]*ps[b,s,r]) * iw[r]
//                   + sum_r pt[b,t,r]*wt_out[r]
//                   + sum_r ps[b,s,r]*ws_out[r] + bias
// Block = 16x16 threads, tile = 32x32 (t,s) pairs, 2x2 micro-tile per thread.
// pt/ps tiles staged in LDS (rows padded to 65 floats: psT column reads would
// otherwise be 16-way bank-conflicted at stride 64).
// ---------------------------------------------------------------------------
__device__ __forceinline__ float fast_silu(float x) {
    // silu(x) = x / (1 + exp(-x));  exp(-x) = exp2(-x * log2(e))
    const float e = __builtin_amdgcn_exp2f(x * -1.44269504088896340736f);
    return x * __builtin_amdgcn_rcpf(1.0f + e);
}

__global__ __launch_bounds__(256)
void pairwise_silu(const float* __restrict__ pt,   // [B*T, RANK]
                   const float* __restrict__ ps,   // [B*S, RANK]
                   const float* __restrict__ wt,   // [RANK]
                   const float* __restrict__ ws,   // [RANK]
                   const float* __restrict__ iw,   // [RANK]
                   const float* __restrict__ bias, // [1]
                   float* __restrict__ out)        // [B, T, S]
{
    __shared__ float sPt[32][RANK + 1];
    __shared__ float sPs[32][RANK + 1];
    __shared__ float sWt[RANK];
    __shared__ float sWs[RANK];
    __shared__ float sIw[RANK];

    const int tx  = threadIdx.x;              // 0..15 -> s
    const int ty  = threadIdx.y;              // 0..15 -> t
    const int tid = ty * 16 + tx;

    const int b  = blockIdx.z;
    const int t0 = blockIdx.y * 32;
    const int s0 = blockIdx.x * 32;

    const float* ptBase = pt + ((size_t)b * TLEN + t0) * RANK;
    const float* psBase = ps + ((size_t)b * SLEN + s0) * RANK;

    // Stage tiles: 32*64 floats each, 256 threads -> 8 floats per thread per tile
    #pragma unroll
    for (int i = tid; i < 32 * RANK; i += 256) {
        const int r   = i & (RANK - 1);
        const int row = i >> 6;
        sPt[row][r] = ptBase[(size_t)row * RANK + r];
        sPs[row][r] = psBase[(size_t)row * RANK + r];
    }
    if (tid < RANK) {
        sWt[tid] = wt[tid];
        sWs[tid] = ws[tid];
        sIw[tid] = iw[tid];
    }
    __syncthreads();

    const int ti0 = 2 * ty, ti1 = 2 * ty + 1;
    const int si0 = 2 * tx, si1 = 2 * tx + 1;

    float acc00 = 0.f, acc01 = 0.f, acc10 = 0.f, acc11 = 0.f;
    float tl0 = 0.f, tl1 = 0.f, sl0 = 0.f, sl1 = 0.f;

    #pragma unroll 4
    for (int r = 0; r < RANK; ++r) {
        const float p0 = sPt[ti0][r];
        const float p1 = sPt[ti1][r];
        const float q0 = sPs[si0][r];
        const float q1 = sPs[si1][r];
        const float w  = sIw[r];
        tl0 = fmaf(p0, sWt[r], tl0);
        tl1 = fmaf(p1, sWt[r], tl1);
        sl0 = fmaf(q0, sWs[r], sl0);
        sl1 = fmaf(q1, sWs[r], sl1);
        acc00 = fmaf(fast_silu(p0 * q0), w, acc00);
        acc01 = fmaf(fast_silu(p0 * q1), w, acc01);
        acc10 = fmaf(fast_silu(p1 * q0), w, acc10);
        acc11 = fmaf(fast_silu(p1 * q1), w, acc11);
    }

    const float bb = *bias;
    float* row0 = out + (((size_t)b * TLEN + (t0 + ti0)) * SLEN + s0 + si0);
    float* row1 = out + (((size_t)b * TLEN + (t0 + ti1)) * SLEN + s0 + si0);
    // s0 + si0 is even -> 8-byte aligned float2 stores
    float2 v0 = make_float2(acc00 + tl0 + sl0 + bb, acc01 + tl0 + sl1 + bb);
    float2 v1 = make_float2(acc10 + tl1 + sl0 + bb, acc11 + tl1 + sl1 + bb);
    *(float2*)row0 = v0;
    *(float2*)row1 = v1;
}

extern "C" void kernel_launch(void* const* d_in, const int* in_sizes, int n_in,
                              void* d_out, int out_size, void* d_ws, size_t ws_size,
                              hipStream_t stream) {
    const float* tgt    = (const float*)d_in[0];  // [B,T,DIM]
    const float* src    = (const float*)d_in[1];  // [B,S,DIM]
    const float* Wt     = (const float*)d_in[2];  // [RANK,DIM]
    const float* Ws     = (const float*)d_in[3];  // [RANK,DIM]
    const float* wt_out = (const float*)d_in[4];  // [RANK]
    const float* ws_out = (const float*)d_in[5];  // [RANK]
    const float* iw     = (const float*)d_in[6];  // [RANK]
    const float* bias   = (const float*)d_in[7];  // [1]
    float* out = (float*)d_out;                   // [B,T,S]

    float* pt = (float*)d_ws;                     // [B*T, RANK] = 512 KB
    float* ps = pt + (size_t)BATCH * TLEN * RANK; // [B*S, RANK] = 512 KB

    // Projections via f32 WMMA: grid = (M/16, RANK/16), 1 wave per 16x16 tile
    dim3 gProj((BATCH * TLEN) / 16, RANK / 16);
    proj_wmma_f32<<<gProj, 32, 0, stream>>>(tgt, Wt, pt);
    proj_wmma_f32<<<gProj, 32, 0, stream>>>(src, Ws, ps);

    // Pairwise tile kernel: grid = (S/32, T/32, B), block = 16x16
    dim3 gPair(SLEN / 32, TLEN / 32, BATCH);
    pairwise_silu<<<gPair, dim3(16, 16), 0, stream>>>(pt, ps, wt_out, ws_out, iw, bias, out);
}